// MCHCGraphSage_69681549410499
// MI455X (gfx1250) — compile-verified
//
#include <hip/hip_runtime.h>
#include <hip/hip_bf16.h>

#define NN 50000      // nodes
#define NE 800000     // edges
#define D  128        // feature dim
#define NL 3          // layers

#define NTILE_NODE (NN / 16)   // 3125 (exact)
#define NTILE_EDGE (NE / 16)   // 50000 (exact)
#define WSTRIDE 68             // LDS k-stride: 68 % 64 == 4 -> conflict-free b64 reads

typedef __attribute__((ext_vector_type(2))) float v2f;
typedef __attribute__((ext_vector_type(8))) float v8f;

// ---------------------------------------------------------------- utilities

__device__ __forceinline__ void atomic_max_f32(float* addr, float val) {
    // monotone float<->int bit trick; mx is initialized to -inf (0xFF800000)
    if (val >= 0.0f) {
        atomicMax((int*)addr, __float_as_int(val));
    } else {
        atomicMin((unsigned int*)addr, __float_as_uint(val));
    }
}

__device__ __forceinline__ float zero_nonfinite(float v) {
    return ((__float_as_uint(v) & 0x7F800000u) == 0x7F800000u) ? 0.0f : v;
}

// ---------------------------------------------------------------- degree

__global__ void deg_kernel(const int* __restrict__ dst, int* __restrict__ deg) {
    int e = blockIdx.x * blockDim.x + threadIdx.x;
    if (e < NE) atomicAdd(&deg[dst[e]], 1);
}

// ------------------------------------------------------ scatter-add (mean num)

__global__ void scatter_add_kernel(const float* __restrict__ x,
                                   const int* __restrict__ src,
                                   const int* __restrict__ dst,
                                   float* __restrict__ agg) {
    unsigned tid = blockIdx.x * blockDim.x + threadIdx.x;
    unsigned e = tid >> 5;
    if (e >= NE) return;
    unsigned part = tid & 31;                       // 32 lanes * float4 = 128
    const float4 v = ((const float4*)(x + (size_t)src[e] * D))[part];
    float* base = agg + (size_t)dst[e] * D + part * 4;
    atomicAdd(base + 0, v.x);
    atomicAdd(base + 1, v.y);
    atomicAdd(base + 2, v.z);
    atomicAdd(base + 3, v.w);
}

// --------------------------------------------------------------- GEMM stage
// Cooperatively stage a 64-row K-slab of W (row-major [K][N]) TRANSPOSED into
// LDS as sWt[n * WSTRIDE + k], then accumulate A[16xK-slab] @ slab into acc[8]
// with V_WMMA_F32_16X16X4_F32. Called by all 256 threads of the block.
// Inner loop: batch all 8 B-fragment DS loads before the 8 WMMAs so DS latency
// overlaps WMMA issue (partial s_wait_dscnt instead of wait-to-zero).

__device__ __forceinline__ void stage_and_mac(const float* __restrict__ W, int kbase,
                                              const float* __restrict__ aRow, float ascl,
                                              float* __restrict__ sWt, v8f* acc) {
    __syncthreads();                                  // protect previous slab's readers
    #pragma unroll
    for (int i = 0; i < 32; ++i) {                    // 64x128 elements / 256 threads
        int idx = threadIdx.x + i * 256;
        int c = idx & 127;                            // column n (coalesced global read)
        int r = idx >> 7;                             // k within slab
        sWt[c * WSTRIDE + r] = W[(kbase + r) * D + c];
    }
    __syncthreads();

    const int lane = threadIdx.x & 31;
    const int m  = lane & 15;
    const int kk = (lane >> 4) << 1;                  // half-wave K pair: 0 or 2

    for (int k = 0; k < 64; k += 4) {
        v2f a = *(const v2f*)(aRow + kbase + k + kk); // contiguous b64 load
        a.x *= ascl;
        a.y *= ascl;
        v2f b[8];
        #pragma unroll
        for (int nt = 0; nt < 8; ++nt) {              // issue all DS loads first
            b[nt] = *(const v2f*)(sWt + (nt * 16 + m) * WSTRIDE + k + kk);
        }
        #pragma unroll
        for (int nt = 0; nt < 8; ++nt) {              // then drain into WMMAs
            acc[nt] = __builtin_amdgcn_wmma_f32_16x16x4_f32(
                false, a, false, b[nt], (short)0, acc[nt], false, false);
        }
    }
}

// ------------------------------------------- fused SAGE layer (dual fp32 WMMA)
// x_out = relu( (agg/max(deg,1)) @ Wm + x @ Ws + bm + bs )
// 8 waves/block; each wave owns a 16-row x 128-col strip; W slabs shared in LDS.

__global__ __launch_bounds__(256)
void sage_layer_kernel(const float* __restrict__ xin,
                       const float* __restrict__ agg,
                       const int*   __restrict__ deg,
                       const float* __restrict__ Wm, const float* __restrict__ bm,
                       const float* __restrict__ Ws, const float* __restrict__ bs,
                       float* __restrict__ xout) {
    __shared__ float sWt[D * WSTRIDE];                // 34.8 KB

    const int wave = threadIdx.x >> 5;
    const int lane = threadIdx.x & 31;
    int tile = blockIdx.x * 8 + wave;
    if (tile >= NTILE_NODE) tile = NTILE_NODE - 1;    // tail waves recompute last tile
    const int row0 = tile * 16;

    const int m = lane & 15;
    const int row = row0 + m;
    const int dg = deg[row];
    const float scl = 1.0f / (float)(dg > 1 ? dg : 1);
    const float* aggRow = agg + (size_t)row * D;
    const float* xRow   = xin + (size_t)row * D;

    v8f acc[8] = {};
    stage_and_mac(Wm, 0,  aggRow, scl,  sWt, acc);
    stage_and_mac(Wm, 64, aggRow, scl,  sWt, acc);
    stage_and_mac(Ws, 0,  xRow,   1.0f, sWt, acc);
    stage_and_mac(Ws, 64, xRow,   1.0f, sWt, acc);

    const int mbase = (lane >> 4) << 3;               // D layout: M = v + 8*(lane>=16)
    #pragma unroll
    for (int nt = 0; nt < 8; ++nt) {
        const int n = nt * 16 + m;
        const float bias = bm[n] + bs[n];
        #pragma unroll
        for (int v = 0; v < 8; ++v) {
            float val = acc[nt][v] + bias;
            xout[(size_t)(row0 + mbase + v) * D + n] = fmaxf(val, 0.0f);
        }
    }
}

// ------------------------------------------------ edge encoder (gathered WMMA)
// ee[e,:] = x[src[e]] @ We1 + x[dst[e]] @ We2 ; one wave = 16 edges x 128 cols.

__global__ __launch_bounds__(256)
void edge_encoder_kernel(const float* __restrict__ x,
                         const int* __restrict__ esrc, const int* __restrict__ edst,
                         const float* __restrict__ We1, const float* __restrict__ We2,
                         float* __restrict__ ee) {
    __shared__ float sWt[D * WSTRIDE];

    const int wave = threadIdx.x >> 5;
    const int lane = threadIdx.x & 31;
    int tile = blockIdx.x * 8 + wave;                 // NTILE_EDGE divisible by 8
    const int e0 = tile * 16;

    const int m = lane & 15;
    const int e = e0 + m;
    const float* sRow = x + (size_t)esrc[e] * D;      // per-lane gathered rows
    const float* dRow = x + (size_t)edst[e] * D;

    v8f acc[8] = {};
    stage_and_mac(We1, 0,  sRow, 1.0f, sWt, acc);
    stage_and_mac(We1, 64, sRow, 1.0f, sWt, acc);
    stage_and_mac(We2, 0,  dRow, 1.0f, sWt, acc);
    stage_and_mac(We2, 64, dRow, 1.0f, sWt, acc);

    const int mbase = (lane >> 4) << 3;
    #pragma unroll
    for (int nt = 0; nt < 8; ++nt) {
        const int n = nt * 16 + m;
        #pragma unroll
        for (int v = 0; v < 8; ++v) {
            ee[(size_t)(e0 + mbase + v) * D + n] = acc[nt][v];
        }
    }
}

// ---------------------------------------------------------------- max scatter

__global__ void mx_init_kernel(unsigned int* __restrict__ mx) {
    size_t i = (size_t)blockIdx.x * blockDim.x + threadIdx.x;
    if (i < (size_t)NE * D) mx[i] = 0xFF800000u;      // -inf
}

__global__ void scatter_max_kernel(const float* __restrict__ ee,
                                   const int* __restrict__ psrc,
                                   const int* __restrict__ pdst,
                                   float* __restrict__ mx) {
    unsigned tid = blockIdx.x * blockDim.x + threadIdx.x;
    unsigned e = tid >> 5;
    if (e >= NE) return;
    unsigned part = tid & 31;
    const float4 v = ((const float4*)(ee + (size_t)psrc[e] * D))[part];
    float* base = mx + (size_t)pdst[e] * D + part * 4;
    atomic_max_f32(base + 0, v.x);
    atomic_max_f32(base + 1, v.y);
    atomic_max_f32(base + 2, v.z);
    atomic_max_f32(base + 3, v.w);
}

// -------------------------------------------------------------- final matvec
// out[e] = mx[e]·W_dl + b_dl + ee[e]·W_dr   (one wave per edge, shuffle reduce)

__global__ __launch_bounds__(256)
void edge_out_kernel(const float* __restrict__ mx, const float* __restrict__ ee,
                     const float* __restrict__ Wdl, const float* __restrict__ bdl,
                     const float* __restrict__ Wdr, float* __restrict__ out) {
    unsigned tid = blockIdx.x * blockDim.x + threadIdx.x;
    unsigned e = tid >> 5;
    if (e >= NE) return;
    int lane = threadIdx.x & 31;
    float4 mv = ((const float4*)(mx + (size_t)e * D))[lane];
    float4 ev = ((const float4*)(ee + (size_t)e * D))[lane];
    float4 wl = ((const float4*)Wdl)[lane];
    float4 wr = ((const float4*)Wdr)[lane];
    mv.x = zero_nonfinite(mv.x); mv.y = zero_nonfinite(mv.y);
    mv.z = zero_nonfinite(mv.z); mv.w = zero_nonfinite(mv.w);
    float p = mv.x * wl.x + mv.y * wl.y + mv.z * wl.z + mv.w * wl.w
            + ev.x * wr.x + ev.y * wr.y + ev.z * wr.z + ev.w * wr.w;
    #pragma unroll
    for (int off = 16; off > 0; off >>= 1) p += __shfl_down(p, off, 32);
    if (lane == 0) out[e] = p + bdl[0];
}

// ---------------------------------------------------------------- launcher

extern "C" void kernel_launch(void* const* d_in, const int* in_sizes, int n_in,
                              void* d_out, int out_size, void* d_ws, size_t ws_size,
                              hipStream_t stream) {
    const float* x      = (const float*)d_in[0];
    const int*   ei     = (const int*)d_in[1];
    // d_in[2] = class_edge_index (unused by reference)
    const int*   pei    = (const int*)d_in[3];
    const float* W_msg  = (const float*)d_in[4];
    const float* b_msg  = (const float*)d_in[5];
    const float* W_self = (const float*)d_in[6];
    const float* b_self = (const float*)d_in[7];
    const float* W_e1   = (const float*)d_in[8];
    const float* W_e2   = (const float*)d_in[9];
    const float* W_dl   = (const float*)d_in[10];
    const float* b_dl   = (const float*)d_in[11];
    const float* W_dr   = (const float*)d_in[12];
    float* out = (float*)d_out;

    const int* src  = ei;             // edge_index[0]
    const int* dst  = ei + NE;        // edge_index[1]
    const int* psrc = pei;
    const int* pdst = pei + NE;

    // workspace carve-out (~896 MB)
    char* ws = (char*)d_ws;
    float* bufA = (float*)ws; ws += (size_t)NN * D * sizeof(float);
    float* bufB = (float*)ws; ws += (size_t)NN * D * sizeof(float);
    float* agg  = (float*)ws; ws += (size_t)NN * D * sizeof(float);
    int*   deg  = (int*)ws;   ws += (size_t)NN * sizeof(int);
    float* ee   = (float*)ws; ws += (size_t)NE * D * sizeof(float);
    float* mx   = (float*)ws; ws += (size_t)NE * D * sizeof(float);

    // degrees (edge list is identical for all layers)
    hipMemsetAsync(deg, 0, (size_t)NN * sizeof(int), stream);
    deg_kernel<<<(NE + 255) / 256, 256, 0, stream>>>(dst, deg);

    const int scatterBlocks = (NE * 32 + 255) / 256;        // 100000
    const int sageBlocks    = (NTILE_NODE + 7) / 8;         // 391
    const int encBlocks     = NTILE_EDGE / 8;               // 6250

    const float* xin = x;
    float* xout = bufA;
    for (int l = 0; l < NL; ++l) {
        hipMemsetAsync(agg, 0, (size_t)NN * D * sizeof(float), stream);
        scatter_add_kernel<<<scatterBlocks, 256, 0, stream>>>(xin, src, dst, agg);
        sage_layer_kernel<<<sageBlocks, 256, 0, stream>>>(
            xin, agg, deg,
            W_msg + (size_t)l * D * D, b_msg + (size_t)l * D,
            W_self + (size_t)l * D * D, b_self + (size_t)l * D, xout);
        xin = xout;
        xout = (xout == bufA) ? bufB : bufA;
    }

    edge_encoder_kernel<<<encBlocks, 256, 0, stream>>>(xin, src, dst, W_e1, W_e2, ee);

    mx_init_kernel<<<(int)(((size_t)NE * D + 255) / 256), 256, 0, stream>>>((unsigned int*)mx);
    scatter_max_kernel<<<scatterBlocks, 256, 0, stream>>>(ee, psrc, pdst, mx);
    edge_out_kernel<<<scatterBlocks, 256, 0, stream>>>(mx, ee, W_dl, b_dl, W_dr, out);
}